// SpringMassSystem_21861383536841
// MI455X (gfx1250) — compile-verified
//
#include <hip/hip_runtime.h>
#include <stdint.h>

// ---------------- problem constants (match reference) ----------------
#define NVERT   10000
#define NSPR    80000
#define NSTEPS  1000
#define F_DT    5e-5f
#define F_DASHPOT 100.0f
#define F_DRAG  3.0f
#define F_GRAVZ -9.8f

// ---------------- launch shape ----------------
#define NWG 64
#define BLK 256
#define SPW ((NSPR + NWG - 1) / NWG)   // 1250 springs per workgroup -> 20000 B LDS

typedef unsigned int u32;
typedef u32 v4u __attribute__((ext_vector_type(4)));
typedef int  v8i __attribute__((ext_vector_type(8)));
typedef int  v4i __attribute__((ext_vector_type(4)));

// Pack spring static data {i1, i2, rest_len, exp(Y)} into 16B records (SoA->AoS),
// and zero the grid-barrier counter for this run.
__global__ void __launch_bounds__(256)
pack_springs(const int* __restrict__ springs, const float* __restrict__ rest,
             const float* __restrict__ springY, uint4* __restrict__ packed,
             u32* __restrict__ bar_cnt) {
  int s = blockIdx.x * blockDim.x + threadIdx.x;
  if (s == 0) *bar_cnt = 0u;
  if (s < NSPR) {
    uint4 q;
    q.x = (u32)springs[2 * s + 0];
    q.y = (u32)springs[2 * s + 1];
    q.z = __float_as_uint(rest[s]);
    q.w = __float_as_uint(expf(springY[s]));
    packed[s] = q;
  }
}

// Monotonic-counter grid barrier (no reset races across 2001 barriers/run).
__device__ __forceinline__ void grid_sync(u32* cnt, u32 target) {
  __syncthreads();
  if (threadIdx.x == 0) {
    __threadfence();  // release all prior global writes
    __hip_atomic_fetch_add(cnt, 1u, __ATOMIC_RELEASE, __HIP_MEMORY_SCOPE_AGENT);
    while (__hip_atomic_load(cnt, __ATOMIC_ACQUIRE, __HIP_MEMORY_SCOPE_AGENT) < target)
      __builtin_amdgcn_s_sleep(2);
    __threadfence();  // acquire: invalidate WGP-local caches
  }
  __syncthreads();
}

__global__ void __launch_bounds__(BLK)
sim(const float* __restrict__ x0, const float* __restrict__ masses,
    const uint4* __restrict__ packed,
    float* __restrict__ x, float* __restrict__ v, float* __restrict__ f,
    u32* bar_cnt, float* __restrict__ out /* [NVERT*3] x, then [NSPR*3] sf */) {
  __shared__ uint4 s_spr[SPW];
  const int tid  = (int)threadIdx.x;
  const int wg   = (int)blockIdx.x;
  const int nthr = (int)(gridDim.x * blockDim.x);
  const int gtid = wg * (int)blockDim.x + tid;

  const int sbase  = wg * SPW;
  const int scount = min(SPW, NSPR - sbase);

  // ---- TDM: async-DMA this WG's spring table into LDS (issued by wave 0) ----
  if (tid < warpSize) {
    u32 lds_addr = (u32)(uintptr_t)&s_spr[0];
    uint64_t ga  = (uint64_t)(uintptr_t)(packed + sbase);
    u32 ndw = (u32)scount * 4u;                 // tile length in dwords
    v4u g0;
    g0[0] = 1u;                                  // count=1 (valid descriptor)
    g0[1] = lds_addr;                            // LDS byte address
    g0[2] = (u32)(ga & 0xFFFFFFFFu);             // global_addr[31:0]
    g0[3] = (u32)((ga >> 32) & 0x01FFFFFFu) | 0x80000000u;  // addr[56:32] | type=2
    v8i g1;
    g1[0] = (int)(2u << 16);                     // data_size = 4 bytes
    g1[1] = (int)((ndw & 0xFFFFu) << 16);        // tensor_dim0[15:0]
    g1[2] = (int)((ndw >> 16) | (1u << 16));     // tensor_dim0[31:16] | tensor_dim1=1
    g1[3] = (int)((ndw & 0xFFFFu) << 16);        // tile_dim0 = ndw
    g1[4] = 1;                                   // tile_dim1=1, tile_dim2=unused
    g1[5] = (int)ndw;                            // tensor_dim0_stride lo
    g1[6] = 0;
    g1[7] = 0;
    v4i gz4 = {0, 0, 0, 0};
    v8i gz8 = {0, 0, 0, 0, 0, 0, 0, 0};
    __builtin_amdgcn_tensor_load_to_lds(g0, g1, gz4, gz4, gz8, 0);
  }

  // ---- state init (overlaps with the TDM transfer) ----
  for (int i = gtid; i < NVERT; i += nthr) {
    float m = masses[i];
    x[3*i+0] = x0[3*i+0]; x[3*i+1] = x0[3*i+1]; x[3*i+2] = x0[3*i+2];
    v[3*i+0] = 0.f;       v[3*i+1] = 0.f;       v[3*i+2] = 0.f;
    f[3*i+0] = 0.f;       f[3*i+1] = 0.f;       f[3*i+2] = m * F_GRAVZ;
  }
  if (tid < warpSize) __builtin_amdgcn_s_wait_tensorcnt(0);

  u32 bar = 0;
  bar += (u32)gridDim.x; grid_sync(bar_cnt, bar);   // LDS + state ready everywhere

  const float damp = expf(-F_DT * F_DRAG);

  for (int step = 0; step < NSTEPS; ++step) {
    const bool last = (step == NSTEPS - 1);

    // ---- phase A: spring + dashpot forces, scatter via f32 atomics ----
    for (int ls = tid; ls < scount; ls += (int)blockDim.x) {
      uint4 q = s_spr[ls];
      int i1 = (int)q.x, i2 = (int)q.y;
      float rl = __uint_as_float(q.z);
      float k  = __uint_as_float(q.w);
      float ax = x[3*i1+0], ay = x[3*i1+1], az = x[3*i1+2];
      float bx = x[3*i2+0], by = x[3*i2+1], bz = x[3*i2+2];
      float dx = bx - ax, dy = by - ay, dz = bz - az;
      float len = sqrtf(dx*dx + dy*dy + dz*dz);
      float inv = 1.0f / len;
      float ex = dx*inv, ey = dy*inv, ez = dz*inv;
      float fs = k * (len / rl - 1.0f);            // elastic magnitude
      float rvx = v[3*i2+0] - v[3*i1+0];
      float rvy = v[3*i2+1] - v[3*i1+1];
      float rvz = v[3*i2+2] - v[3*i1+2];
      float vrel = rvx*ex + rvy*ey + rvz*ez;
      float ft = fs + F_DASHPOT * vrel;
      float px = ft*ex, py = ft*ey, pz = ft*ez;
      atomicAdd(&f[3*i1+0],  px); atomicAdd(&f[3*i1+1],  py); atomicAdd(&f[3*i1+2],  pz);
      atomicAdd(&f[3*i2+0], -px); atomicAdd(&f[3*i2+1], -py); atomicAdd(&f[3*i2+2], -pz);
      if (last) {                                   // elastic part only, per reference
        int gs = sbase + ls;
        out[3*NVERT + 3*gs + 0] = fs * ex;
        out[3*NVERT + 3*gs + 1] = fs * ey;
        out[3*NVERT + 3*gs + 2] = fs * ez;
      }
    }
    bar += (u32)gridDim.x; grid_sync(bar_cnt, bar);

    // ---- phase B: symplectic-Euler integrate + ground clamp + force reset ----
    for (int i = gtid; i < NVERT; i += nthr) {
      float m = masses[i];
      float s = F_DT / m;
      float vx = (v[3*i+0] + f[3*i+0]*s) * damp;
      float vy = (v[3*i+1] + f[3*i+1]*s) * damp;
      float vz = (v[3*i+2] + f[3*i+2]*s) * damp;
      float nx = x[3*i+0] + F_DT*vx;
      float ny = x[3*i+1] + F_DT*vy;
      float nz = x[3*i+2] + F_DT*vz;
      nz = fmaxf(nz, 0.0f);
      if (nz == 0.0f) vz = 0.0f;
      x[3*i+0] = nx; x[3*i+1] = ny; x[3*i+2] = nz;
      v[3*i+0] = vx; v[3*i+1] = vy; v[3*i+2] = vz;
      f[3*i+0] = 0.f; f[3*i+1] = 0.f; f[3*i+2] = m * F_GRAVZ;  // next-step init
      if (last) { out[3*i+0] = nx; out[3*i+1] = ny; out[3*i+2] = nz; }
    }
    bar += (u32)gridDim.x; grid_sync(bar_cnt, bar);
  }
}

extern "C" void kernel_launch(void* const* d_in, const int* in_sizes, int n_in,
                              void* d_out, int out_size, void* d_ws, size_t ws_size,
                              hipStream_t stream) {
  const float* x0      = (const float*)d_in[0];   // init_vertices (10000,3) f32
  const int*   springs = (const int*)  d_in[1];   // springs (80000,2) i32
  const float* rest    = (const float*)d_in[2];   // rest_lengths (80000,) f32
  const float* masses  = (const float*)d_in[3];   // masses (10000,) f32
  const float* springY = (const float*)d_in[4];   // spring_Y (80000,) f32
  float* out = (float*)d_out;                     // [30000 x] ++ [240000 spring_forces]

  char* ws = (char*)d_ws;
  uint4* packed = (uint4*)ws;                         // 80000 * 16 B
  float* x = (float*)(ws + (size_t)NSPR * 16);        // 30000 f32
  float* v = x + (size_t)NVERT * 3;                   // 30000 f32
  float* f = v + (size_t)NVERT * 3;                   // 30000 f32
  u32* bar_cnt = (u32*)(f + (size_t)NVERT * 3);

  pack_springs<<<(NSPR + 255) / 256, 256, 0, stream>>>(springs, rest, springY,
                                                       packed, bar_cnt);
  sim<<<NWG, BLK, 0, stream>>>(x0, masses, packed, x, v, f, bar_cnt, out);
}